// NGCF_8461085573271
// MI455X (gfx1250) — compile-verified
//
#include <hip/hip_runtime.h>
#include <math.h>

// ---------------------------------------------------------------------------
// NGCF forward (BPR + reg loss) for MI455X / gfx1250 (wave32, WMMA).
// Strategy:
//   - Build CSR once (histogram + scan + scatter) -> atomic-free SpMM.
//   - SpMM: 1 wave per row, float4 per lane, ego rows live in 192MB L2.
//   - Fused dual GEMM (side@Wg + (ego*side)@Wb + biases) with
//     V_WMMA_F32_16X16X4_F32, leaky-relu, in-place ego update.
//   - Scores accumulated per layer from unnormalized ego via norms+dots,
//     so the [N,512] concat matrix is never materialized.
// ---------------------------------------------------------------------------

typedef float v2f __attribute__((ext_vector_type(2)));
typedef float v8f __attribute__((ext_vector_type(8)));

#define DIM 128
#define TS 132          // padded LDS tile stride (conflict-free b64 reads)
#define NEG_SLOPE 0.2f
#define REG_LAMBDA 1e-4f
#define EPSN 1e-12f

__device__ __forceinline__ float wave_reduce_add(float v) {
#pragma unroll
  for (int off = 16; off > 0; off >>= 1) v += __shfl_xor(v, off, 32);
  return v;
}

// ---------------- setup kernels --------------------------------------------

__global__ __launch_bounds__(256) void zero_init(int* cnt, float* sp, float* sn,
                                                 float* out, int N, int B) {
  int idx = blockIdx.x * 256 + threadIdx.x;
  if (idx < N) cnt[idx] = 0;
  if (idx < B) { sp[idx] = 0.f; sn[idx] = 0.f; }
  if (idx < 2) out[idx] = 0.f;
}

__global__ __launch_bounds__(256) void init_ego(const float* __restrict__ ue,
                                                const float* __restrict__ ie,
                                                float* __restrict__ ego,
                                                int U, int N) {
  int idx = blockIdx.x * 256 + threadIdx.x;       // one float4 per thread
  int total4 = N * (DIM / 4);
  if (idx >= total4) return;
  int u4 = U * (DIM / 4);
  float4 v = (idx < u4) ? ((const float4*)ue)[idx] : ((const float4*)ie)[idx - u4];
  ((float4*)ego)[idx] = v;
}

__global__ __launch_bounds__(256) void hist_rows(const int* __restrict__ er,
                                                 int* __restrict__ cnt, int NNZ) {
  int idx = blockIdx.x * 256 + threadIdx.x;
  if (idx < NNZ) atomicAdd(&cnt[er[idx]], 1);
}

__global__ __launch_bounds__(256) void scan_block(const int* __restrict__ cnt,
                                                  int* __restrict__ rp,
                                                  int* __restrict__ bsum, int N) {
  __shared__ int sh[256];
  int idx = blockIdx.x * 256 + threadIdx.x;
  int v = (idx < N) ? cnt[idx] : 0;
  sh[threadIdx.x] = v;
  __syncthreads();
  int val = v;
#pragma unroll
  for (int off = 1; off < 256; off <<= 1) {
    int other = (threadIdx.x >= off) ? sh[threadIdx.x - off] : 0;
    __syncthreads();
    val += other;
    sh[threadIdx.x] = val;
    __syncthreads();
  }
  if (idx < N) rp[idx] = val - v;                 // block-local exclusive
  if (threadIdx.x == 255) bsum[blockIdx.x] = val; // block total
}

__global__ __launch_bounds__(1024) void scan_sums(int* __restrict__ bsum, int nb) {
  __shared__ int sh[1024];
  int v = (threadIdx.x < nb) ? bsum[threadIdx.x] : 0;
  sh[threadIdx.x] = v;
  __syncthreads();
  int val = v;
#pragma unroll
  for (int off = 1; off < 1024; off <<= 1) {
    int other = (threadIdx.x >= off) ? sh[threadIdx.x - off] : 0;
    __syncthreads();
    val += other;
    sh[threadIdx.x] = val;
    __syncthreads();
  }
  if (threadIdx.x < nb) bsum[threadIdx.x] = val - v;  // exclusive
}

__global__ __launch_bounds__(256) void scan_add(int* __restrict__ rp,
                                                const int* __restrict__ bsum,
                                                int* __restrict__ cursor,
                                                int N, int NNZ) {
  int idx = blockIdx.x * 256 + threadIdx.x;
  if (idx < N) {
    int v = rp[idx] + bsum[idx >> 8];
    rp[idx] = v;
    cursor[idx] = v;
  }
  if (idx == N) rp[N] = NNZ;
}

__global__ __launch_bounds__(256) void scatter_edges(const int* __restrict__ er,
                                                     const int* __restrict__ ec,
                                                     const float* __restrict__ ev,
                                                     int* __restrict__ cursor,
                                                     int* __restrict__ colsS,
                                                     float* __restrict__ valsS,
                                                     int NNZ) {
  int idx = blockIdx.x * 256 + threadIdx.x;
  if (idx >= NNZ) return;
  int r = er[idx];
  int p = atomicAdd(&cursor[r], 1);
  colsS[p] = ec[idx];
  valsS[p] = ev[idx];
}

// ---------------- SpMM: side = A @ ego (CSR, 1 wave / row) ------------------

__global__ __launch_bounds__(256) void spmm_csr(const int* __restrict__ rp,
                                                const int* __restrict__ cols,
                                                const float* __restrict__ vals,
                                                const float* __restrict__ ego,
                                                float* __restrict__ side, int N) {
  int wave = (blockIdx.x * 256 + threadIdx.x) >> 5;
  int lane = threadIdx.x & 31;
  if (wave >= N) return;
  int beg = rp[wave], end = rp[wave + 1];
  float4 acc = make_float4(0.f, 0.f, 0.f, 0.f);
  for (int e = beg; e < end; ++e) {
    int c = cols[e];
    float v = vals[e];
    if (e + 1 < end)  // pull next gathered row toward L0/L2 (global_prefetch_b8)
      __builtin_prefetch(ego + (size_t)cols[e + 1] * DIM + lane * 4, 0, 3);
    const float4 x = *(const float4*)(ego + (size_t)c * DIM + lane * 4);
    acc.x = fmaf(v, x.x, acc.x);
    acc.y = fmaf(v, x.y, acc.y);
    acc.z = fmaf(v, x.z, acc.z);
    acc.w = fmaf(v, x.w, acc.w);
  }
  *(float4*)(side + (size_t)wave * DIM + lane * 4) = acc;
}

// ---------------- fused dual GEMM + leaky-relu (WMMA f32 16x16x4) -----------
// ego_new = lrelu(side@Wg + bg + (ego*side)@Wb + bb), written in place.

__global__ __launch_bounds__(256) void gemm_fused(const float* __restrict__ side,
                                                  float* __restrict__ ego,
                                                  const float* __restrict__ Wg,
                                                  const float* __restrict__ bg,
                                                  const float* __restrict__ Wb,
                                                  const float* __restrict__ bb,
                                                  int N) {
  __shared__ float sA[16 * TS];   // side tile
  __shared__ float pA[16 * TS];   // (ego*side) tile
  const int r0 = blockIdx.x * 16;
  const int t  = threadIdx.x;

  // stage 16x128 A-tiles into LDS (float4 loads, 2 per thread)
  for (int q = t; q < 16 * 32; q += 256) {
    int r = q >> 5, c = (q & 31) << 2;
    int row = r0 + r;
    float4 s = make_float4(0.f, 0.f, 0.f, 0.f);
    float4 e = make_float4(0.f, 0.f, 0.f, 0.f);
    if (row < N) {
      s = *(const float4*)(side + (size_t)row * DIM + c);
      e = *(const float4*)(ego  + (size_t)row * DIM + c);
    }
    float* sp = sA + r * TS + c;
    float* pp = pA + r * TS + c;
    sp[0] = s.x; sp[1] = s.y; sp[2] = s.z; sp[3] = s.w;
    pp[0] = e.x * s.x; pp[1] = e.y * s.y; pp[2] = e.z * s.z; pp[3] = e.w * s.w;
  }
  __syncthreads();

  const int wave = t >> 5;
  const int lane = t & 31;
  const int half = lane >> 4;   // 0: K=k0..k0+1 | 1: K=k0+2..k0+3  (A layout)
  const int tcol = lane & 15;   // M for A-frag, N for B/C-frag
  const int n0   = wave * 16;   // this wave's 16 output columns

  v8f acc;
  const float bias = bg[n0 + tcol] + bb[n0 + tcol];
#pragma unroll
  for (int r = 0; r < 8; ++r) acc[r] = bias;

  // GEMM 1: side @ Wg  (32 chained K-steps of 4)
#pragma unroll 4
  for (int k0 = 0; k0 < DIM; k0 += 4) {
    v2f a, b;
    const float* ap = sA + tcol * TS + k0 + 2 * half;
    a.x = ap[0]; a.y = ap[1];
    const float* bp = Wg + (size_t)(k0 + 2 * half) * DIM + n0 + tcol;
    b.x = bp[0]; b.y = bp[DIM];
    acc = __builtin_amdgcn_wmma_f32_16x16x4_f32(false, a, false, b,
                                                (short)0, acc, false, false);
  }
  // GEMM 2: (ego*side) @ Wb, accumulated into same C
#pragma unroll 4
  for (int k0 = 0; k0 < DIM; k0 += 4) {
    v2f a, b;
    const float* ap = pA + tcol * TS + k0 + 2 * half;
    a.x = ap[0]; a.y = ap[1];
    const float* bp = Wb + (size_t)(k0 + 2 * half) * DIM + n0 + tcol;
    b.x = bp[0]; b.y = bp[DIM];
    acc = __builtin_amdgcn_wmma_f32_16x16x4_f32(false, a, false, b,
                                                (short)0, acc, false, false);
  }

  // leaky-relu + in-place writeback (C layout: M = r + 8*half, N = n0 + tcol)
#pragma unroll
  for (int r = 0; r < 8; ++r) {
    int row = r0 + r + 8 * half;
    if (row < N) {
      float v = acc[r];
      v = (v > 0.f) ? v : NEG_SLOPE * v;
      ego[(size_t)row * DIM + n0 + tcol] = v;
    }
  }
}

// ---------------- score accumulation ---------------------------------------

// layer-0 segment: raw embeddings (no normalization) + reg term
__global__ __launch_bounds__(256) void score_layer0(const int* __restrict__ user,
                                                    const int* __restrict__ pos,
                                                    const int* __restrict__ neg,
                                                    const float* __restrict__ ue,
                                                    const float* __restrict__ ie,
                                                    float* __restrict__ sp,
                                                    float* __restrict__ sn,
                                                    float* __restrict__ out,
                                                    int B) {
  int b    = (blockIdx.x * 256 + threadIdx.x) >> 5;
  int lane = threadIdx.x & 31;
  if (b >= B) return;
  const float4 u = *(const float4*)(ue + (size_t)user[b] * DIM + lane * 4);
  const float4 p = *(const float4*)(ie + (size_t)pos[b]  * DIM + lane * 4);
  const float4 n = *(const float4*)(ie + (size_t)neg[b]  * DIM + lane * 4);
  float dp = u.x * p.x + u.y * p.y + u.z * p.z + u.w * p.w;
  float dn = u.x * n.x + u.y * n.y + u.z * n.z + u.w * n.w;
  float rg = p.x * p.x + p.y * p.y + p.z * p.z + p.w * p.w +
             n.x * n.x + n.y * n.y + n.z * n.z + n.w * n.w;
  dp = wave_reduce_add(dp);
  dn = wave_reduce_add(dn);
  rg = wave_reduce_add(rg);
  if (lane == 0) {
    sp[b] += dp;
    sn[b] += dn;
    atomicAdd(&out[1], rg * (REG_LAMBDA * 0.5f / (float)B));
  }
}

// layer-l segment: normalized dots computed from unnormalized ego rows
__global__ __launch_bounds__(256) void score_layer(const int* __restrict__ user,
                                                   const int* __restrict__ pos,
                                                   const int* __restrict__ neg,
                                                   const float* __restrict__ ego,
                                                   float* __restrict__ sp,
                                                   float* __restrict__ sn,
                                                   int U, int B) {
  int b    = (blockIdx.x * 256 + threadIdx.x) >> 5;
  int lane = threadIdx.x & 31;
  if (b >= B) return;
  const float4 u = *(const float4*)(ego + (size_t)user[b] * DIM + lane * 4);
  const float4 p = *(const float4*)(ego + (size_t)(U + pos[b]) * DIM + lane * 4);
  const float4 n = *(const float4*)(ego + (size_t)(U + neg[b]) * DIM + lane * 4);
  float up = u.x * p.x + u.y * p.y + u.z * p.z + u.w * p.w;
  float un = u.x * n.x + u.y * n.y + u.z * n.z + u.w * n.w;
  float uu = u.x * u.x + u.y * u.y + u.z * u.z + u.w * u.w;
  float pp = p.x * p.x + p.y * p.y + p.z * p.z + p.w * p.w;
  float nn = n.x * n.x + n.y * n.y + n.z * n.z + n.w * n.w;
  up = wave_reduce_add(up);
  un = wave_reduce_add(un);
  uu = wave_reduce_add(uu);
  pp = wave_reduce_add(pp);
  nn = wave_reduce_add(nn);
  if (lane == 0) {
    float nu = fmaxf(sqrtf(uu), EPSN);
    float np = fmaxf(sqrtf(pp), EPSN);
    float nv = fmaxf(sqrtf(nn), EPSN);
    sp[b] += up / (nu * np);
    sn[b] += un / (nu * nv);
  }
}

__global__ __launch_bounds__(256) void bpr_loss(const float* __restrict__ sp,
                                                const float* __restrict__ sn,
                                                float* __restrict__ out, int B) {
  __shared__ float sh[256];
  int idx = blockIdx.x * 256 + threadIdx.x;
  float v = 0.f;
  if (idx < B) {
    float x = sn[idx] - sp[idx];                 // softplus(neg - pos)
    v = fmaxf(x, 0.f) + log1pf(expf(-fabsf(x)));
  }
  sh[threadIdx.x] = v;
  __syncthreads();
#pragma unroll
  for (int off = 128; off > 0; off >>= 1) {
    if (threadIdx.x < off) sh[threadIdx.x] += sh[threadIdx.x + off];
    __syncthreads();
  }
  if (threadIdx.x == 0) atomicAdd(&out[0], sh[0] / (float)B);
}

// ---------------- launcher --------------------------------------------------

extern "C" void kernel_launch(void* const* d_in, const int* in_sizes, int n_in,
                              void* d_out, int out_size, void* d_ws, size_t ws_size,
                              hipStream_t stream) {
  const int*   user  = (const int*)  d_in[0];
  const int*   pos   = (const int*)  d_in[1];
  const int*   neg   = (const int*)  d_in[2];
  const int*   erow  = (const int*)  d_in[3];
  const int*   ecol  = (const int*)  d_in[4];
  const float* eval  = (const float*)d_in[5];
  const float* uemb  = (const float*)d_in[6];
  const float* iemb  = (const float*)d_in[7];
  const float* Wgcn  = (const float*)d_in[8];
  const float* bgcn  = (const float*)d_in[9];
  const float* Wbi   = (const float*)d_in[10];
  const float* bbi   = (const float*)d_in[11];
  float* out = (float*)d_out;

  const int B   = in_sizes[0];
  const int NNZ = in_sizes[3];
  const int U   = in_sizes[6] / DIM;
  const int I   = in_sizes[7] / DIM;
  const int N   = U + I;
  const int L   = in_sizes[8] / (DIM * DIM);

  // bump-allocate workspace (256B aligned regions)
  char* wp = (char*)d_ws;
  auto alloc = [&](size_t bytes) -> void* {
    void* r = (void*)wp;
    wp += (bytes + 255) & ~(size_t)255;
    return r;
  };
  int*   cnt    = (int*)  alloc((size_t)N * 4);
  int*   bsum   = (int*)  alloc(1024 * 4);
  int*   rp     = (int*)  alloc((size_t)(N + 1) * 4);
  int*   cursor = (int*)  alloc((size_t)N * 4);
  int*   colsS  = (int*)  alloc((size_t)NNZ * 4);
  float* valsS  = (float*)alloc((size_t)NNZ * 4);
  float* ego    = (float*)alloc((size_t)N * DIM * 4);
  float* side   = (float*)alloc((size_t)N * DIM * 4);
  float* sPos   = (float*)alloc((size_t)B * 4);
  float* sNeg   = (float*)alloc((size_t)B * 4);

  const int nbScan = (N + 255) / 256;

  zero_init<<<dim3((N + 255) / 256), dim3(256), 0, stream>>>(cnt, sPos, sNeg, out, N, B);
  init_ego<<<dim3((N * (DIM / 4) + 255) / 256), dim3(256), 0, stream>>>(uemb, iemb, ego, U, N);

  // CSR build
  hist_rows<<<dim3((NNZ + 255) / 256), dim3(256), 0, stream>>>(erow, cnt, NNZ);
  scan_block<<<dim3(nbScan), dim3(256), 0, stream>>>(cnt, rp, bsum, N);
  scan_sums<<<dim3(1), dim3(1024), 0, stream>>>(bsum, nbScan);
  scan_add<<<dim3((N + 1 + 255) / 256), dim3(256), 0, stream>>>(rp, bsum, cursor, N, NNZ);
  scatter_edges<<<dim3((NNZ + 255) / 256), dim3(256), 0, stream>>>(erow, ecol, eval,
                                                                   cursor, colsS, valsS, NNZ);

  // layer-0 scores + reg term from raw embeddings
  score_layer0<<<dim3((B * 32 + 255) / 256), dim3(256), 0, stream>>>(user, pos, neg, uemb,
                                                                     iemb, sPos, sNeg, out, B);

  const dim3 spmmGrid((N * 32 + 255) / 256);
  const dim3 gemmGrid((N + 15) / 16);
  const dim3 scoreGrid((B * 32 + 255) / 256);
  for (int l = 0; l < L; ++l) {
    spmm_csr<<<spmmGrid, dim3(256), 0, stream>>>(rp, colsS, valsS, ego, side, N);
    gemm_fused<<<gemmGrid, dim3(256), 0, stream>>>(side, ego,
                                                   Wgcn + (size_t)l * DIM * DIM,
                                                   bgcn + (size_t)l * DIM,
                                                   Wbi  + (size_t)l * DIM * DIM,
                                                   bbi  + (size_t)l * DIM, N);
    score_layer<<<scoreGrid, dim3(256), 0, stream>>>(user, pos, neg, ego, sPos, sNeg, U, B);
  }

  bpr_loss<<<dim3((B + 255) / 256), dim3(256), 0, stream>>>(sPos, sNeg, out, B);
}